// GraphGenericNetwork_19954418057369
// MI455X (gfx1250) — compile-verified
//
#include <hip/hip_runtime.h>
#include <math.h>

typedef __attribute__((ext_vector_type(2))) float v2f;
typedef __attribute__((ext_vector_type(8))) float v8f;

#define N_NODES   21
#define NODE_FEAT 128
#define N_ACT     256
#define NTHREADS  256
#define NWAVES    8

// y = relu(x @ W + b) via fp32 WMMA 16x16x4.
// x: zero-padded vector in LDS = row 0 of the 16xK A operand (rows 1..15 zero).
// W: [K, N] row-major global. Each wave owns T contiguous 16-wide N tiles and
// keeps T independent accumulator chains so WMMAs pipeline without RAW stalls.
template <int T>
__device__ __forceinline__ void fc_layer_wmma(
    const float* __restrict__ W, const float* __restrict__ bias,
    const float* __restrict__ xLDS, float* __restrict__ y,
    int K, int N, int wave, int lane)
{
    const int  n_in_tile = lane & 15;
    const int  hi        = lane >> 4;          // lanes 16..31 carry K+2,K+3 slices
    const bool arow      = (n_in_tile == 0);   // only row M=0 of A is live
    const int  kmain     = K & ~3;
    const float* Wbase   = W + wave * T * 16 + n_in_tile;

    v8f c[T];
#pragma unroll
    for (int t = 0; t < T; ++t) c[t] = (v8f){};

    for (int k0 = 0; k0 < kmain; k0 += 4) {
        const int ka = k0 + (hi << 1);                 // even -> b64 LDS load ok
        v2f xv = *(const v2f*)(xLDS + ka);             // unconditional broadcast
        v2f a;
        a[0] = arow ? xv[0] : 0.0f;                    // branchless v_cndmask
        a[1] = arow ? xv[1] : 0.0f;
        const float* wp = Wbase + (size_t)ka * N;
#pragma unroll
        for (int t = 0; t < T; ++t) {
            v2f b;
            b[0] = wp[t * 16];                         // row ka
            b[1] = wp[t * 16 + N];                     // row ka+1
            c[t] = __builtin_amdgcn_wmma_f32_16x16x4_f32(
                       false, a, false, b, (short)0, c[t], false, false);
        }
    }
    if (kmain < K) {                                   // K=441 tail, branchless
        const int ka = kmain + (hi << 1);
        const int kb = ka + 1;
        const int kca = (ka < K) ? ka : (K - 1);       // clamp -> valid address
        const int kcb = (kb < K) ? kb : (K - 1);
        v2f xv = *(const v2f*)(xLDS + ka);             // xLDS is zero-padded
        v2f a;
        a[0] = arow ? xv[0] : 0.0f;
        a[1] = arow ? xv[1] : 0.0f;
#pragma unroll
        for (int t = 0; t < T; ++t) {
            float bx = Wbase[(size_t)kca * N + t * 16];  // unconditional loads
            float by = Wbase[(size_t)kcb * N + t * 16];
            v2f b;
            b[0] = (ka < K) ? bx : 0.0f;               // v_cndmask, no exec ops
            b[1] = (kb < K) ? by : 0.0f;
            c[t] = __builtin_amdgcn_wmma_f32_16x16x4_f32(
                       false, a, false, b, (short)0, c[t], false, false);
        }
    }
    if (lane < 16) {                                   // row M=0 -> c[t][0], lanes 0..15
#pragma unroll
        for (int t = 0; t < T; ++t) {
            int n = wave * T * 16 + t * 16 + lane;
            float r = c[t][0] + bias[n];
            y[n] = fmaxf(r, 0.0f);
        }
    }
}

__global__ __launch_bounds__(NTHREADS)
void gcn_mlp_fused(const float* __restrict__ state, const int* __restrict__ ei,
                   const float* __restrict__ W1,  const float* __restrict__ b1,
                   const float* __restrict__ W2,  const float* __restrict__ b2,
                   const float* __restrict__ Wf1, const float* __restrict__ bf1,
                   const float* __restrict__ Wf2, const float* __restrict__ bf2,
                   const float* __restrict__ Wf3, const float* __restrict__ bf3,
                   float* __restrict__ out)
{
    __shared__ float s_state[N_NODES * NODE_FEAT];         // state[0], 21x128
    __shared__ float s_A[448];                             // (normalized) adjacency
    __shared__ float s_dinv[24];
    __shared__ float bufA[448];
    __shared__ __align__(16) float bufB[448];              // ends as v[441], zero-padded
    __shared__ __align__(16) float s_x1[512];
    __shared__ __align__(16) float s_x2[512];

    const int tid  = threadIdx.x;
    const int lane = tid & 31;
    const int wave = tid >> 5;

    // ---- stage 0: stage state[0] into LDS, zero adjacency ----
    for (int i = tid; i < N_NODES * NODE_FEAT; i += NTHREADS) s_state[i] = state[i];
    for (int i = tid; i < 448; i += NTHREADS) s_A[i] = 0.0f;

    // Warm L2/WGP$ with the FC weights (global_prefetch_b8: no LOADcnt, free
    // overlap with the GCN stage below; this kernel is pure latency-bound).
    {
        const char* p1 = (const char*)Wf1;
        const char* p2 = (const char*)Wf2;
        const char* p3 = (const char*)Wf3;
        const size_t n1 = (size_t)441 * 512 * 4;
        const size_t n2 = (size_t)512 * 512 * 4;
        const size_t n3 = (size_t)512 * 256 * 4;
        for (size_t o = (size_t)tid * 128; o < n1; o += (size_t)NTHREADS * 128)
            __builtin_prefetch(p1 + o, 0, 3);
        for (size_t o = (size_t)tid * 128; o < n2; o += (size_t)NTHREADS * 128)
            __builtin_prefetch(p2 + o, 0, 3);
        for (size_t o = (size_t)tid * 128; o < n3; o += (size_t)NTHREADS * 128)
            __builtin_prefetch(p3 + o, 0, 3);
    }
    __syncthreads();

    // ---- stage 1: A = D^-1/2 (Adj + I) D^-1/2 ----
    if (tid < 128) {
        int r = ei[tid], c = ei[128 + tid];
        atomicAdd(&s_A[r * 21 + c], 1.0f);                 // ds_add_f32 scatter
    }
    __syncthreads();
    if (tid < 21) s_A[tid * 21 + tid] += 1.0f;             // self loops
    __syncthreads();
    if (tid < 21) {                                        // deg = column sums
        float d = 0.0f;
        for (int r = 0; r < 21; ++r) d += s_A[r * 21 + tid];
        s_dinv[tid] = (d > 0.0f) ? (1.0f / sqrtf(d)) : 0.0f;
    }
    __syncthreads();
    for (int i = tid; i < 441; i += NTHREADS) {
        int r = i / 21, c = i - r * 21;
        s_A[i] *= s_dinv[r] * s_dinv[c];
    }
    __syncthreads();

    // ---- stage 2: GCN layers on batch element 0 only ----
    // h1 = X @ W1
    for (int i = tid; i < 441; i += NTHREADS) {
        int n = i / 21, h = i - n * 21;
        float acc = 0.0f;
        const float* xr = &s_state[n * NODE_FEAT];
        for (int f = 0; f < NODE_FEAT; ++f) acc = fmaf(xr[f], W1[f * 21 + h], acc);
        bufA[i] = acc;
    }
    __syncthreads();
    // h2 = A @ h1 + b1
    for (int i = tid; i < 441; i += NTHREADS) {
        int n = i / 21, h = i - n * 21;
        float acc = b1[h];
        for (int j = 0; j < 21; ++j) acc = fmaf(s_A[n * 21 + j], bufA[j * 21 + h], acc);
        bufB[i] = acc;
    }
    __syncthreads();
    // h3 = h2 @ W2
    for (int i = tid; i < 441; i += NTHREADS) {
        int n = i / 21, h = i - n * 21;
        float acc = 0.0f;
        for (int f = 0; f < 21; ++f) acc = fmaf(bufB[n * 21 + f], W2[f * 21 + h], acc);
        bufA[i] = acc;
    }
    __syncthreads();
    // v = A @ h3 + b2, zero-padded to 448 for the WMMA K tail
    for (int i = tid; i < 448; i += NTHREADS) {
        if (i < 441) {
            int n = i / 21, h = i - n * 21;
            float acc = b2[h];
            for (int j = 0; j < 21; ++j) acc = fmaf(s_A[n * 21 + j], bufA[j * 21 + h], acc);
            bufB[i] = acc;
        } else {
            bufB[i] = 0.0f;
        }
    }
    __syncthreads();

    // ---- stage 3: MLP head via fp32 WMMA ----
    fc_layer_wmma<4>(Wf1, bf1, bufB, s_x1, 441, 512, wave, lane);  // 32 tiles
    __syncthreads();
    fc_layer_wmma<4>(Wf2, bf2, s_x1, s_x2, 512, 512, wave, lane);  // 32 tiles
    __syncthreads();
    fc_layer_wmma<2>(Wf3, bf3, s_x2, out, 512, N_ACT, wave, lane); // 16 tiles
}

extern "C" void kernel_launch(void* const* d_in, const int* in_sizes, int n_in,
                              void* d_out, int out_size, void* d_ws, size_t ws_size,
                              hipStream_t stream) {
    const float* state = (const float*)d_in[0];
    const int*   ei    = (const int*)  d_in[1];
    const float* W1    = (const float*)d_in[2];
    const float* b1    = (const float*)d_in[3];
    const float* W2    = (const float*)d_in[4];
    const float* b2    = (const float*)d_in[5];
    const float* Wf1   = (const float*)d_in[6];
    const float* bf1   = (const float*)d_in[7];
    const float* Wf2   = (const float*)d_in[8];
    const float* bf2   = (const float*)d_in[9];
    const float* Wf3   = (const float*)d_in[10];
    const float* bf3   = (const float*)d_in[11];
    float* out = (float*)d_out;

    // Only state[0] matters (reference takes reshape(B,-1)[0]); the whole
    // network fits one workgroup -> single fused launch, everything in LDS.
    gcn_mlp_fused<<<1, NTHREADS, 0, stream>>>(state, ei, W1, b1, W2, b2,
                                              Wf1, bf1, Wf2, bf2, Wf3, bf3, out);
}